// MambaTimeEmbedding_81698867905023
// MI455X (gfx1250) — compile-verified
//
#include <hip/hip_runtime.h>
#include <hip/hip_bf16.h>
#include <math.h>

// Problem constants (from reference)
#define BB     2
#define LL     2048
#define CIN    64
#define HH     256
#define NLAYER 4
#define II     512      // 2*H
#define SS     16
#define RR     16
#define BLR    (BB*LL)  // 4096 rows
#define NC     32       // scan chunks
#define CLEN   (LL/NC)  // 64 per chunk

typedef float v2f __attribute__((ext_vector_type(2)));
typedef float v8f __attribute__((ext_vector_type(8)));

// ---------------------------------------------------------------------------
// FP32 WMMA 16x16x4 helpers (wave32).
// A (16x4): lane&15 = row M, (lane>>4)*2+j = K element j of the v2f.
// B (4x16): lane&15 = col N, (lane>>4)*2+j = K row   (W stored NxK row-major).
// C/D (16x16): VGPR j -> (M = j + (lane>>4)*8, N = lane&15).
// ---------------------------------------------------------------------------
__device__ __forceinline__ v8f wmma4(v2f a, v2f b, v8f c) {
  return __builtin_amdgcn_wmma_f32_16x16x4_f32(false, a, false, b, (short)0, c,
                                               false, false);
}

// Software-pipelined accumulate: D(16 x 16*NT) = A(16 x K) * W(16*NT x K)^T.
// Fragments for k-step k+4 are loaded before the WMMAs for k-step k issue,
// so the per-iteration wait only covers loads issued a full iteration earlier
// (keeps ~NT+1 loads in flight under the matrix ops instead of stalling).
template <int NT>
__device__ __forceinline__ void gemm_acc(const float* __restrict__ A, int lda,
                                         const float* __restrict__ W, int ldw,
                                         int m0, int n0, int Kdim, int lane,
                                         v8f* c) {
  const float* pa = A + (size_t)(m0 + (lane & 15)) * lda + ((lane >> 4) << 1);
  const float* pw = W + (size_t)(n0 + (lane & 15)) * ldw + ((lane >> 4) << 1);
  v2f a, b[NT];
  a.x = pa[0]; a.y = pa[1];
#pragma unroll
  for (int t = 0; t < NT; ++t) {
    const float* q = pw + (size_t)(16 * t) * ldw;
    b[t].x = q[0]; b[t].y = q[1];
  }
  int k0 = 0;
  for (; k0 + 4 < Kdim; k0 += 4) {
    v2f an, bn[NT];
    const float* qa = pa + k0 + 4;
    an.x = qa[0]; an.y = qa[1];
#pragma unroll
    for (int t = 0; t < NT; ++t) {
      const float* q = pw + (size_t)(16 * t) * ldw + k0 + 4;
      bn[t].x = q[0]; bn[t].y = q[1];
    }
#pragma unroll
    for (int t = 0; t < NT; ++t) c[t] = wmma4(a, b[t], c[t]);
    a = an;
#pragma unroll
    for (int t = 0; t < NT; ++t) b[t] = bn[t];
  }
#pragma unroll
  for (int t = 0; t < NT; ++t) c[t] = wmma4(a, b[t], c[t]);
}

// ---------------------------------------------------------------------------
// 1) h = x @ proj_w^T + proj_b + positional encoding     (M=4096,N=256,K=64)
// ---------------------------------------------------------------------------
__global__ __launch_bounds__(256) void k_proj_pos(
    const float* __restrict__ x, const float* __restrict__ pw,
    const float* __restrict__ pb, const float* __restrict__ tsp,
    float* __restrict__ h) {
  int lane = threadIdx.x & 31, wave = threadIdx.x >> 5;
  int m0 = blockIdx.y * 128 + wave * 16;
  int n0 = blockIdx.x * 64;
  v8f c[4] = {};
  gemm_acc<4>(x, CIN, pw, CIN, m0, n0, CIN, lane, c);
  int hi8 = (lane >> 4) << 3, ncol = lane & 15;
#pragma unroll
  for (int t = 0; t < 4; ++t) {
    int n = n0 + 16 * t + ncol;
    int idx = (n < HH / 2) ? n : n - HH / 2;
    // period = 2^(-10 + 10*idx/127); ang = 2*pi*rel/period
    float w = 6.28318530718f / exp2f(10.f * (float)idx / 127.f - 10.f);
#pragma unroll
    for (int j = 0; j < 8; ++j) {
      int row = m0 + j + hi8;
      int b = row >> 11;
      float rel = tsp[row] - tsp[b << 11];
      float ang = w * rel;
      float pe = (n < HH / 2) ? __sinf(ang) : __cosf(ang);
      h[(size_t)row * HH + n] = c[t][j] + pb[n] + pe;
    }
  }
}

// ---------------------------------------------------------------------------
// RMSNorm over H=256 (one 256-thread block per row); also used for final norm
// ---------------------------------------------------------------------------
__global__ __launch_bounds__(256) void k_rmsnorm(const float* __restrict__ h,
                                                 const float* __restrict__ w,
                                                 float* __restrict__ o) {
  __shared__ float red[256];
  int row = blockIdx.x, col = threadIdx.x;
  float v = h[(size_t)row * HH + col];
  red[col] = v * v;
  __syncthreads();
  for (int s = 128; s > 0; s >>= 1) {
    if (col < s) red[col] += red[col + s];
    __syncthreads();
  }
  float scale = rsqrtf(red[0] * (1.0f / HH) + 1e-5f);
  o[(size_t)row * HH + col] = v * scale * w[col];
}

// ---------------------------------------------------------------------------
// 2) proj = hn @ in_w^T  (M=4096, N=1024, K=256). cols [0,512)=u*mask, rest=gate
// ---------------------------------------------------------------------------
__global__ __launch_bounds__(256) void k_inproj(
    const float* __restrict__ hn, const float* __restrict__ W,
    const int* __restrict__ mask, float* __restrict__ proj) {
  int lane = threadIdx.x & 31, wave = threadIdx.x >> 5;
  int m0 = blockIdx.y * 128 + wave * 16;
  int n0 = blockIdx.x * 64;
  v8f c[4] = {};
  gemm_acc<4>(hn, HH, W, HH, m0, n0, HH, lane, c);
  int hi8 = (lane >> 4) << 3, ncol = lane & 15;
#pragma unroll
  for (int t = 0; t < 4; ++t) {
    int n = n0 + 16 * t + ncol;
#pragma unroll
    for (int j = 0; j < 8; ++j) {
      int row = m0 + j + hi8;
      float v = c[t][j];
      if (n < II) v *= (float)mask[row];
      proj[(size_t)row * (2 * II) + n] = v;
    }
  }
}

// ---------------------------------------------------------------------------
// 3) depthwise causal conv (K=4) + bias + SiLU + mask -> uc (BL x 512)
// ---------------------------------------------------------------------------
__global__ __launch_bounds__(256) void k_conv(
    const float* __restrict__ proj, const float* __restrict__ cw,
    const float* __restrict__ cb, const int* __restrict__ mask,
    float* __restrict__ uc) {
  int idx = blockIdx.x * blockDim.x + threadIdx.x;   // over BL*II
  int i = idx & (II - 1);
  int bl = idx >> 9;
  int l = bl & (LL - 1);
  float s = cb[i];
#pragma unroll
  for (int k = 0; k < 4; ++k) {
    int lk = l + k - 3;
    if (lk >= 0) s += proj[(size_t)(bl + k - 3) * (2 * II) + i] * cw[i * 4 + k];
  }
  float sv = s / (1.f + __expf(-s));
  uc[(size_t)bl * II + i] = sv * (float)mask[bl];
}

// ---------------------------------------------------------------------------
// 4) dbc = uc @ x_proj_w^T  (M=4096, N=48, K=512); wave does 16x48
// ---------------------------------------------------------------------------
__global__ __launch_bounds__(256) void k_dbc(const float* __restrict__ uc,
                                             const float* __restrict__ xw,
                                             float* __restrict__ dbc) {
  int lane = threadIdx.x & 31, wave = threadIdx.x >> 5;
  int m0 = blockIdx.x * 128 + wave * 16;
  v8f c[3] = {};
  gemm_acc<3>(uc, II, xw, II, m0, 0, II, lane, c);
  int hi8 = (lane >> 4) << 3, ncol = lane & 15;
#pragma unroll
  for (int t = 0; t < 3; ++t) {
    int n = 16 * t + ncol;
#pragma unroll
    for (int j = 0; j < 8; ++j) {
      int row = m0 + j + hi8;
      dbc[(size_t)row * 48 + n] = c[t][j];
    }
  }
}

// ---------------------------------------------------------------------------
// 5) dt = softplus(ts @ dt_proj_w^T + dtb)  (M=4096, N=512, K=16)
//    ts = dbc[:, 0:16] (lda=48)
// ---------------------------------------------------------------------------
__global__ __launch_bounds__(256) void k_dt(const float* __restrict__ dbc,
                                            const float* __restrict__ dtw,
                                            const float* __restrict__ dtb,
                                            float* __restrict__ dt) {
  int lane = threadIdx.x & 31, wave = threadIdx.x >> 5;
  int m0 = blockIdx.y * 128 + wave * 16;
  int n0 = blockIdx.x * 64;
  v8f c[4] = {};
  gemm_acc<4>(dbc, 48, dtw, RR, m0, n0, RR, lane, c);
  int hi8 = (lane >> 4) << 3, ncol = lane & 15;
#pragma unroll
  for (int t = 0; t < 4; ++t) {
    int n = n0 + 16 * t + ncol;
    float bv = dtb[n];
#pragma unroll
    for (int j = 0; j < 8; ++j) {
      int row = m0 + j + hi8;
      float xv = c[t][j] + bv;
      float sp = (xv > 20.f) ? xv : log1pf(__expf(xv));
      dt[(size_t)row * II + n] = sp;
    }
  }
}

// ---------------------------------------------------------------------------
// SSM scan, 3-phase chunked over L. state_t = dA_t*state_{t-1} + dt*B*u
// ---------------------------------------------------------------------------
__global__ __launch_bounds__(256) void k_scan1(
    const float* __restrict__ dt, const float* __restrict__ uc,
    const float* __restrict__ dbc, const float* __restrict__ alog,
    float* __restrict__ sP, float* __restrict__ sS) {
  int e = blockIdx.x * blockDim.x + threadIdx.x;   // B*II*NC
  int i = e & (II - 1);
  int q = e >> 9;                  // ch*B + b
  int b = q & (BB - 1);
  int ch = q >> 1;
  float A_[SS];
#pragma unroll
  for (int s = 0; s < SS; ++s) A_[s] = -__expf(alog[i * SS + s]);
  float pr[SS], st[SS];
#pragma unroll
  for (int s = 0; s < SS; ++s) { pr[s] = 1.f; st[s] = 0.f; }
  int l0 = ch * CLEN;
  for (int t = 0; t < CLEN; ++t) {
    int bl = b * LL + l0 + t;
    float dtv = dt[(size_t)bl * II + i];
    float uv = uc[(size_t)bl * II + i];
    float du = dtv * uv;
    const float* dr = dbc + (size_t)bl * 48 + SS;  // B part
#pragma unroll
    for (int s = 0; s < SS; ++s) {
      float dA = __expf(dtv * A_[s]);
      pr[s] *= dA;
      st[s] = st[s] * dA + du * dr[s];
    }
  }
  size_t base = ((size_t)(b * II + i) * NC + ch) * SS;
#pragma unroll
  for (int s = 0; s < SS; ++s) { sP[base + s] = pr[s]; sS[base + s] = st[s]; }
}

__global__ __launch_bounds__(256) void k_scan2(const float* __restrict__ sP,
                                               const float* __restrict__ sS,
                                               float* __restrict__ sI) {
  int e = blockIdx.x * blockDim.x + threadIdx.x;   // B*II channels
  float carry[SS] = {};
  for (int c = 0; c < NC; ++c) {
    size_t base = ((size_t)e * NC + c) * SS;
#pragma unroll
    for (int s = 0; s < SS; ++s) {
      sI[base + s] = carry[s];
      carry[s] = sP[base + s] * carry[s] + sS[base + s];
    }
  }
}

__global__ __launch_bounds__(256) void k_scan3(
    const float* __restrict__ dt, const float* __restrict__ uc,
    const float* __restrict__ dbc, const float* __restrict__ alog,
    const float* __restrict__ Dp, const float* __restrict__ proj,
    const float* __restrict__ sI, float* __restrict__ y) {
  int e = blockIdx.x * blockDim.x + threadIdx.x;
  int i = e & (II - 1);
  int q = e >> 9;
  int b = q & (BB - 1);
  int ch = q >> 1;
  float A_[SS];
#pragma unroll
  for (int s = 0; s < SS; ++s) A_[s] = -__expf(alog[i * SS + s]);
  float st[SS];
  size_t ibase = ((size_t)(b * II + i) * NC + ch) * SS;
#pragma unroll
  for (int s = 0; s < SS; ++s) st[s] = sI[ibase + s];
  float Dv = Dp[i];
  int l0 = ch * CLEN;
  for (int t = 0; t < CLEN; ++t) {
    int bl = b * LL + l0 + t;
    float dtv = dt[(size_t)bl * II + i];
    float uv = uc[(size_t)bl * II + i];
    float du = dtv * uv;
    const float* dr = dbc + (size_t)bl * 48;
    float acc = 0.f;
#pragma unroll
    for (int s = 0; s < SS; ++s) {
      float dA = __expf(dtv * A_[s]);
      st[s] = st[s] * dA + du * dr[SS + s];       // B part
      acc += st[s] * dr[2 * SS + s];              // C part
    }
    float yv = acc + Dv * uv;
    float g = proj[(size_t)bl * (2 * II) + II + i];
    yv *= g / (1.f + __expf(-g));                  // * silu(gate)
    y[(size_t)bl * II + i] = yv;
  }
}

// ---------------------------------------------------------------------------
// 6) h += y @ out_proj_w^T  (M=4096, N=256, K=512), fused residual add
// ---------------------------------------------------------------------------
__global__ __launch_bounds__(256) void k_outproj(const float* __restrict__ y,
                                                 const float* __restrict__ W,
                                                 float* __restrict__ h) {
  int lane = threadIdx.x & 31, wave = threadIdx.x >> 5;
  int m0 = blockIdx.y * 128 + wave * 16;
  int n0 = blockIdx.x * 64;
  v8f c[4] = {};
  gemm_acc<4>(y, II, W, II, m0, n0, II, lane, c);
  int hi8 = (lane >> 4) << 3, ncol = lane & 15;
#pragma unroll
  for (int t = 0; t < 4; ++t) {
    int n = n0 + 16 * t + ncol;
#pragma unroll
    for (int j = 0; j < 8; ++j) {
      int row = m0 + j + hi8;
      h[(size_t)row * HH + n] += c[t][j];
    }
  }
}

// ---------------------------------------------------------------------------
extern "C" void kernel_launch(void* const* d_in, const int* in_sizes, int n_in,
                              void* d_out, int out_size, void* d_ws,
                              size_t ws_size, hipStream_t stream) {
  const float* x      = (const float*)d_in[0];
  const float* tsp    = (const float*)d_in[1];
  const int*   mask   = (const int*)  d_in[2];
  const float* proj_w = (const float*)d_in[3];
  const float* proj_b = (const float*)d_in[4];
  const float* norm_w = (const float*)d_in[5];
  const float* in_w   = (const float*)d_in[6];
  const float* conv_w = (const float*)d_in[7];
  const float* conv_b = (const float*)d_in[8];
  const float* xw     = (const float*)d_in[9];
  const float* dtw    = (const float*)d_in[10];
  const float* dtbp   = (const float*)d_in[11];
  const float* alog   = (const float*)d_in[12];
  const float* Dp     = (const float*)d_in[13];
  const float* ow     = (const float*)d_in[14];
  const float* nfw    = (const float*)d_in[15];

  float* ws   = (float*)d_ws;
  float* h    = ws;                              // BL*H
  float* hn   = h    + (size_t)BLR * HH;         // BL*H
  float* proj = hn   + (size_t)BLR * HH;         // BL*2I
  float* uc   = proj + (size_t)BLR * 2 * II;     // BL*I
  float* dbc  = uc   + (size_t)BLR * II;         // BL*48
  float* dtB  = dbc  + (size_t)BLR * 48;         // BL*I
  float* yb   = dtB  + (size_t)BLR * II;         // BL*I
  float* sP   = yb   + (size_t)BLR * II;         // B*I*NC*S
  float* sS   = sP   + (size_t)BB * II * NC * SS;
  float* sI   = sS   + (size_t)BB * II * NC * SS;

  k_proj_pos<<<dim3(HH / 64, BLR / 128), 256, 0, stream>>>(x, proj_w, proj_b,
                                                           tsp, h);
  for (int l = 0; l < NLAYER; ++l) {
    k_rmsnorm<<<BLR, HH, 0, stream>>>(h, norm_w + (size_t)l * HH, hn);
    k_inproj<<<dim3(2 * II / 64, BLR / 128), 256, 0, stream>>>(
        hn, in_w + (size_t)l * 2 * II * HH, mask, proj);
    k_conv<<<(BLR * II) / 256, 256, 0, stream>>>(
        proj, conv_w + (size_t)l * II * 4, conv_b + (size_t)l * II, mask, uc);
    k_dbc<<<BLR / 128, 256, 0, stream>>>(uc, xw + (size_t)l * 48 * II, dbc);
    k_dt<<<dim3(II / 64, BLR / 128), 256, 0, stream>>>(
        dbc, dtw + (size_t)l * II * RR, dtbp + (size_t)l * II, dtB);
    k_scan1<<<(BB * II * NC) / 256, 256, 0, stream>>>(
        dtB, uc, dbc, alog + (size_t)l * II * SS, sP, sS);
    k_scan2<<<(BB * II) / 256, 256, 0, stream>>>(sP, sS, sI);
    k_scan3<<<(BB * II * NC) / 256, 256, 0, stream>>>(
        dtB, uc, dbc, alog + (size_t)l * II * SS, Dp + (size_t)l * II, proj,
        sI, yb);
    k_outproj<<<dim3(HH / 64, BLR / 128), 256, 0, stream>>>(
        yb, ow + (size_t)l * HH * II, h);
  }
  k_rmsnorm<<<BLR, HH, 0, stream>>>(h, nfw, (float*)d_out);
}